// AZConv2d_3152505995453
// MI455X (gfx1250) — compile-verified
//
#include <hip/hip_runtime.h>
#include <math.h>

typedef float v2f __attribute__((ext_vector_type(2)));
typedef float v8f __attribute__((ext_vector_type(8)));

#define CIN   64
#define RGATE 4
#define HH    128
#define WWID  128
#define LPIX  (HH * WWID)
#define COUT  128
#define KRC   256      // RGATE * CIN
#define TW    16       // tile width  (pixels)
#define TH    2        // tile height (pixels)

// ---------------------------------------------------------------------------
// Kernel 1: mu = softmax_r( x . gate_w + gate_b )   -> ws  (B, 4, L)  fp32
// ---------------------------------------------------------------------------
__global__ __launch_bounds__(256) void mu_kernel(const float* __restrict__ x,
                                                 const float* __restrict__ gate_w,
                                                 const float* __restrict__ gate_b,
                                                 float* __restrict__ mu) {
  int g = blockIdx.x * 256 + threadIdx.x;      // pixel over B*L
  if (g >= 8 * LPIX) return;
  int b  = g >> 14;                            // / 16384
  int hw = g & (LPIX - 1);
  const float* xb = x + ((size_t)b * CIN) * LPIX + hw;

  float a0 = gate_b[0], a1 = gate_b[1], a2 = gate_b[2], a3 = gate_b[3];
  #pragma unroll 8
  for (int c = 0; c < CIN; ++c) {
    float v = xb[(size_t)c * LPIX];            // coalesced across lanes
    a0 = fmaf(v, gate_w[0 * CIN + c], a0);     // gate_w reads are wave-uniform
    a1 = fmaf(v, gate_w[1 * CIN + c], a1);
    a2 = fmaf(v, gate_w[2 * CIN + c], a2);
    a3 = fmaf(v, gate_w[3 * CIN + c], a3);
  }
  float m  = fmaxf(fmaxf(a0, a1), fmaxf(a2, a3));
  float e0 = expf(a0 - m), e1 = expf(a1 - m), e2 = expf(a2 - m), e3 = expf(a3 - m);
  float inv = 1.0f / (e0 + e1 + e2 + e3);
  float* mb = mu + ((size_t)b * RGATE) * LPIX + hw;
  mb[0 * LPIX] = e0 * inv;
  mb[1 * LPIX] = e1 * inv;
  mb[2 * LPIX] = e2 * inv;
  mb[3 * LPIX] = e3 * inv;
}

// ---------------------------------------------------------------------------
// Kernel 2: fused alpha -> agg (LDS) -> WMMA fp32 GEMM with pw_w -> out
// One workgroup = 16x2 pixel tile; 8 waves x (16 outch x 16 pix) x 2 rows.
// ---------------------------------------------------------------------------
__global__ __launch_bounds__(256) void azconv_kernel(const float* __restrict__ x,
                                                     const float* __restrict__ mu,
                                                     const float* __restrict__ theta,
                                                     const float* __restrict__ rsu,
                                                     const float* __restrict__ rss,
                                                     const float* __restrict__ pw_w,
                                                     const float* __restrict__ pw_b,
                                                     float* __restrict__ out) {
  __shared__ float s_x[CIN][TH + 2][18];       // x halo  (c, h0-1..h0+2, w0-1..w0+16)
  __shared__ float s_mu[RGATE][TH + 2][18];    // mu halo
  __shared__ float s_kern[RGATE][9];           // gaussian kernel
  __shared__ float s_alpha[RGATE * 9][TW * TH];// normalized alpha per pixel
  __shared__ float s_agg[TH][257][TW];         // agg per pixel-row; 257 pads banks

  const int t   = threadIdx.x;
  const int blk = blockIdx.x;                  // b*512 + ht*8 + wt
  const int wt  = blk & 7;
  const int ht  = (blk >> 3) & 63;
  const int b   = blk >> 9;
  const int w0  = wt * TW;
  const int h0  = ht * TH;

  // ---- stage x halo -------------------------------------------------------
  for (int e = t; e < CIN * (TH + 2) * 18; e += 256) {
    int c   = e / ((TH + 2) * 18);
    int rem = e - c * ((TH + 2) * 18);
    int dyi = rem / 18;
    int col = rem - dyi * 18;
    int hh  = h0 + dyi - 1;
    int ww  = w0 + col - 1;
    float v = 0.0f;
    if (hh >= 0 && hh < HH && ww >= 0 && ww < WWID)
      v = x[(((size_t)b * CIN + c) * HH + hh) * WWID + ww];
    s_x[c][dyi][col] = v;
  }
  // ---- stage mu halo ------------------------------------------------------
  for (int e = t; e < RGATE * (TH + 2) * 18; e += 256) {
    int r   = e / ((TH + 2) * 18);
    int rem = e - r * ((TH + 2) * 18);
    int dyi = rem / 18;
    int col = rem - dyi * 18;
    int hh  = h0 + dyi - 1;
    int ww  = w0 + col - 1;
    float v = 0.0f;
    if (hh >= 0 && hh < HH && ww >= 0 && ww < WWID)
      v = mu[(((size_t)b * RGATE + r) * HH + hh) * WWID + ww];
    s_mu[r][dyi][col] = v;
  }
  // ---- tiny gaussian kernel (36 values) -----------------------------------
  if (t < RGATE * 9) {
    int r = t / 9, s = t - r * 9;
    float dy = (float)(s / 3 - 1);
    float dx = (float)(s % 3 - 1);
    float th = theta[r];
    float ct = cosf(th), st = sinf(th);
    float pu =  ct * dx + st * dy;
    float ps = -st * dx + ct * dy;
    float su = log1pf(expf(rsu[r])) + 1e-4f;   // softplus + eps
    float ss = log1pf(expf(rss[r])) + 1e-4f;
    s_kern[r][s] = expf(-(pu * pu) / (su * su) - (ps * ps) / (ss * ss));
  }
  __syncthreads();

  // ---- alpha per pixel (normalized over r,s); one thread per pixel --------
  if (t < TW * TH) {
    int p  = t;
    int py = p >> 4;                           // 0..1
    int px = p & 15;
    float a[36];
    float sum = 0.0f;
    #pragma unroll
    for (int r = 0; r < RGATE; ++r) {
      float mc = s_mu[r][py + 1][px + 1];      // center tap
      #pragma unroll
      for (int s = 0; s < 9; ++s) {
        int dyi = s / 3, dxi = s % 3;
        float v = mc * s_mu[r][py + dyi][px + dxi] * s_kern[r][s];
        a[r * 9 + s] = v;
        sum += v;
      }
    }
    float inv = 1.0f / fmaxf(sum, 1e-8f);
    #pragma unroll
    for (int i = 0; i < 36; ++i) s_alpha[i][p] = a[i] * inv;
  }
  __syncthreads();

  // ---- agg[pt][rc][pl] = sum_s alpha[r,s,p] * x_un[c,s,p] -----------------
  // Each thread owns one pixel (pt,pl) and 32 of the 256 rc entries.
  {
    int pl = t & 15;                           // pixel column
    int pt = (t >> 4) & 1;                     // pixel row
    int g  = t >> 5;                           // 0..7 -> c offset
    int p  = pt * 16 + pl;
    #pragma unroll
    for (int r = 0; r < RGATE; ++r) {
      float a0 = s_alpha[r * 9 + 0][p], a1 = s_alpha[r * 9 + 1][p];
      float a2 = s_alpha[r * 9 + 2][p], a3 = s_alpha[r * 9 + 3][p];
      float a4 = s_alpha[r * 9 + 4][p], a5 = s_alpha[r * 9 + 5][p];
      float a6 = s_alpha[r * 9 + 6][p], a7 = s_alpha[r * 9 + 7][p];
      float a8 = s_alpha[r * 9 + 8][p];
      #pragma unroll
      for (int j = 0; j < 8; ++j) {
        int c = j * 8 + g;
        float acc =      a0 * s_x[c][pt + 0][pl + 0];
        acc = fmaf(a1, s_x[c][pt + 0][pl + 1], acc);
        acc = fmaf(a2, s_x[c][pt + 0][pl + 2], acc);
        acc = fmaf(a3, s_x[c][pt + 1][pl + 0], acc);
        acc = fmaf(a4, s_x[c][pt + 1][pl + 1], acc);
        acc = fmaf(a5, s_x[c][pt + 1][pl + 2], acc);
        acc = fmaf(a6, s_x[c][pt + 2][pl + 0], acc);
        acc = fmaf(a7, s_x[c][pt + 2][pl + 1], acc);
        acc = fmaf(a8, s_x[c][pt + 2][pl + 2], acc);
        s_agg[pt][r * 64 + c][pl] = acc;
      }
    }
  }
  __syncthreads();

  // ---- WMMA fp32 GEMM: one A fragment feeds two independent chains --------
  {
    const int lane = t & 31;
    const int wv   = t >> 5;                   // 0..7 -> out-channel tile
    const int o0   = wv * 16;
    const int m    = lane & 15;                // A row / B column (pixel)
    const int hi   = lane >> 4;                // half-wave selects K pair
    v8f acc0 = {0.f, 0.f, 0.f, 0.f, 0.f, 0.f, 0.f, 0.f};
    v8f acc1 = {0.f, 0.f, 0.f, 0.f, 0.f, 0.f, 0.f, 0.f};

    const float* arow = pw_w + (size_t)(o0 + m) * KRC;
    for (int kk = 0; kk < KRC / 4; ++kk) {
      int kb = kk * 4 + hi * 2;                // lanes 0-15: K0/K1, 16-31: K2/K3
      v2f afrag = *(const v2f*)(arow + kb);    // 8B aligned (kb even)
      v2f b0, b1;
      b0.x = s_agg[0][kb + 0][m];
      b0.y = s_agg[0][kb + 1][m];
      b1.x = s_agg[1][kb + 0][m];
      b1.y = s_agg[1][kb + 1][m];
      acc0 = __builtin_amdgcn_wmma_f32_16x16x4_f32(
          false, afrag, false, b0, (short)0, acc0, false, false);
      acc1 = __builtin_amdgcn_wmma_f32_16x16x4_f32(
          false, afrag, false, b1, (short)0, acc1, false, false);
    }

    // D layout: vgpr i -> M = i (lanes 0-15) / i+8 (lanes 16-31), N = lane&15
    const int w = w0 + m;
    #pragma unroll
    for (int i = 0; i < 8; ++i) {
      int o = o0 + i + hi * 8;
      float bias = pw_b[o];
      out[(((size_t)b * COUT + o) * HH + (h0 + 0)) * WWID + w] = acc0[i] + bias;
      out[(((size_t)b * COUT + o) * HH + (h0 + 1)) * WWID + w] = acc1[i] + bias;
    }
  }
}

// ---------------------------------------------------------------------------
extern "C" void kernel_launch(void* const* d_in, const int* in_sizes, int n_in,
                              void* d_out, int out_size, void* d_ws, size_t ws_size,
                              hipStream_t stream) {
  const float* x      = (const float*)d_in[0];
  const float* gate_w = (const float*)d_in[1];
  const float* gate_b = (const float*)d_in[2];
  const float* theta  = (const float*)d_in[3];
  const float* rsu    = (const float*)d_in[4];
  const float* rss    = (const float*)d_in[5];
  const float* pw_w   = (const float*)d_in[6];
  const float* pw_b   = (const float*)d_in[7];
  float* out = (float*)d_out;
  float* mu  = (float*)d_ws;                   // B*4*L floats = 2 MB scratch

  mu_kernel<<<(8 * LPIX) / 256, 256, 0, stream>>>(x, gate_w, gate_b, mu);
  azconv_kernel<<<8 * (HH / TH) * (WWID / TW), 256, 0, stream>>>(
      x, mu, theta, rsu, rss, pw_w, pw_b, out);
}